// GNNEncoder_43138651521238
// MI455X (gfx1250) — compile-verified
//
#include <hip/hip_runtime.h>
#include <hip/hip_bf16.h>

typedef __attribute__((ext_vector_type(2))) float v2f;
typedef __attribute__((ext_vector_type(8))) float v8f;

#define N_NODES 100000
#define N_EDGES 1000000
#define D_FEAT  64
#define BN_EPS  1e-5f
#define NEG_SLOPE 0.01f

// ---------------------------------------------------------------------------
// Scatter-add message passing: agg[dst,f] += feat[src,f] * w[e]
// One thread per (edge, 4-feature chunk): b128 feature reads, edge metadata
// amortized over 16 threads. The 25.6MB accumulator is L2-resident (192MB L2)
// so the f32 atomics resolve on-chip.
// ---------------------------------------------------------------------------
__global__ __launch_bounds__(256) void scatter_kernel(
    const float* __restrict__ feat,
    const long long* __restrict__ ei,   // [2, N_EDGES] int64
    const float* __restrict__ ew,
    float* __restrict__ agg)
{
    int t = blockIdx.x * 256 + threadIdx.x;   // grid sized exactly: N_EDGES*16
    int e = t >> 4;                           // 16 chunks of 4 feats per edge
    int c = (t & 15) << 2;                    // feature base
    long long s = ei[e];
    long long d = ei[N_EDGES + e];
    float w = ew[e];
    float4 v = *(const float4*)(feat + (size_t)s * D_FEAT + c);
    float* dst = agg + (size_t)d * D_FEAT + c;
    atomicAdd(dst + 0, v.x * w);
    atomicAdd(dst + 1, v.y * w);
    atomicAdd(dst + 2, v.z * w);
    atomicAdd(dst + 3, v.w * w);
}

// ---------------------------------------------------------------------------
// WMMA fp32 GEMM: out[M,DOUT] = A[M,64] @ W[64,DOUT] + bias, fused per-column
// sum / sum-of-squares for BatchNorm. One wave -> one 16x16 C tile via
// V_WMMA_F32_16X16X4_F32, 16 K-steps. W staged in LDS (compile-time DOUT ->
// ds_load_b32 with immediate offsets, no per-iteration address math).
// ---------------------------------------------------------------------------
template<int DOUT>
__global__ __launch_bounds__(256) void gemm_wmma_kernel(
    const float* __restrict__ A,     // [N_NODES, 64]
    const float* __restrict__ W,     // [64, DOUT] row-major
    const float* __restrict__ bias,  // [DOUT]
    float* __restrict__ out,         // [N_NODES, DOUT]
    float* __restrict__ colsum,      // [DOUT] or null
    float* __restrict__ colsumsq)    // [DOUT] or null
{
    __shared__ float Wl[64 * DOUT];

    // Cooperative stage of W into LDS (before any wave-level early exit so the
    // barrier is reached by every wave in the block).
    {
        constexpr int NV4 = (64 * DOUT) / (256 * 4);   // 4 for DOUT=64, 2 for 32
        const float4* Wv  = (const float4*)W;
        float4*       Wlv = (float4*)Wl;
        #pragma unroll
        for (int i = 0; i < NV4; ++i)
            Wlv[threadIdx.x + i * 256] = Wv[threadIdx.x + i * 256];
    }
    __syncthreads();

    const int lane   = threadIdx.x & 31;
    const int wid    = threadIdx.x >> 5;
    const int ntiles = DOUT >> 4;
    const int total  = (N_NODES / 16) * ntiles;        // 100000 % 16 == 0

    int tile = blockIdx.x * 8 + wid;
    if (tile >= total) return;                         // wave-uniform branch
    int mtile = tile / ntiles;
    int ntile = tile - mtile * ntiles;

    const int lrow = lane & 15;
    const int half = lane >> 4;                        // K-half of A/B fragments
    const int n    = ntile * 16 + lrow;                // output column of lane
    const int arow = mtile * 16 + lrow;                // A row held by lane

    const float* Arow = A + (size_t)arow * 64 + half * 2;
    const float* Bcol = Wl + half * 2 * DOUT + n;

    v8f c = {};
    #pragma unroll
    for (int ks = 0; ks < 16; ++ks) {
        // A fragment: 16x4 fp32 -> lane holds A[arow][k0], A[arow][k0+1]
        v2f a = *(const v2f*)(Arow + ks * 4);
        // B fragment (mirror): lane holds W[k0][n], W[k0+1][n] from LDS
        v2f b;
        b.x = Bcol[(ks * 4 + 0) * DOUT];
        b.y = Bcol[(ks * 4 + 1) * DOUT];
#if __has_builtin(__builtin_amdgcn_wmma_f32_16x16x4_f32)
        c = __builtin_amdgcn_wmma_f32_16x16x4_f32(
                /*neg_a=*/false, a, /*neg_b=*/false, b,
                /*c_mod=*/(short)0, c, /*reuse_a=*/false, /*reuse_b=*/false);
#else
        #pragma unroll
        for (int r = 0; r < 8; ++r) c[r] += a.x * b.x + a.y * b.y; // placeholder
#endif
    }

    const float bn = bias[n];
    float* orow = out + (size_t)(mtile * 16 + half * 8) * DOUT + n;
    float s = 0.f, ss = 0.f;
    #pragma unroll
    for (int r = 0; r < 8; ++r) {                      // C layout: VGPR r -> row r/r+8
        float v = c[r] + bn;
        orow[r * DOUT] = v;
        s  += v;
        ss += v * v;
    }
    if (colsum) {
        atomicAdd(&colsum[n], s);
        atomicAdd(&colsumsq[n], ss);
    }
}

// ---------------------------------------------------------------------------
// Fold column sums into BN scale/shift:  y = x*scale[f] + shift[f]
// ---------------------------------------------------------------------------
__global__ void bn_finalize_kernel(
    const float* __restrict__ colsum, const float* __restrict__ colsumsq,
    const float* __restrict__ gamma,  const float* __restrict__ beta,
    float* __restrict__ scale, float* __restrict__ shift)
{
    int f = threadIdx.x;
    const float invN = 1.0f / (float)N_NODES;
    float mean = colsum[f] * invN;
    float var  = colsumsq[f] * invN - mean * mean;     // population var (jnp.var)
    float sc   = gamma[f] * rsqrtf(var + BN_EPS);
    scale[f] = sc;
    shift[f] = beta[f] - mean * sc;
}

// ---------------------------------------------------------------------------
// Apply BN affine + LeakyReLU in place, float4-vectorized (b128 load/store)
// ---------------------------------------------------------------------------
__global__ __launch_bounds__(256) void bn_act_kernel(
    float4* __restrict__ h,
    const float4* __restrict__ scale4, const float4* __restrict__ shift4)
{
    int idx = blockIdx.x * 256 + threadIdx.x;          // grid sized exactly NF/4
    int f4  = idx & 15;                                // feature-chunk within 64
    float4 v  = h[idx];
    float4 sc = scale4[f4];
    float4 sh = shift4[f4];
    v.x = v.x * sc.x + sh.x; v.x = (v.x >= 0.f) ? v.x : NEG_SLOPE * v.x;
    v.y = v.y * sc.y + sh.y; v.y = (v.y >= 0.f) ? v.y : NEG_SLOPE * v.y;
    v.z = v.z * sc.z + sh.z; v.z = (v.z >= 0.f) ? v.z : NEG_SLOPE * v.z;
    v.w = v.w * sc.w + sh.w; v.w = (v.w >= 0.f) ? v.w : NEG_SLOPE * v.w;
    h[idx] = v;
}

// ---------------------------------------------------------------------------
extern "C" void kernel_launch(void* const* d_in, const int* in_sizes, int n_in,
                              void* d_out, int out_size, void* d_ws, size_t ws_size,
                              hipStream_t stream)
{
    const float*     x   = (const float*)d_in[0];
    const long long* ei  = (const long long*)d_in[1];
    const float*     ew  = (const float*)d_in[2];
    // d_in[3] = batch (unused: single graph)
    const float* W1 = (const float*)d_in[4];
    const float* b1 = (const float*)d_in[5];
    const float* g1 = (const float*)d_in[6];
    const float* be1= (const float*)d_in[7];
    const float* W2 = (const float*)d_in[8];
    const float* b2 = (const float*)d_in[9];
    const float* g2 = (const float*)d_in[10];
    const float* be2= (const float*)d_in[11];
    const float* W3 = (const float*)d_in[12];
    const float* b3 = (const float*)d_in[13];
    float* out = (float*)d_out;

    const size_t NF = (size_t)N_NODES * D_FEAT;
    float* bufA   = (float*)d_ws;          // aggregation buffer
    float* bufB   = bufA + NF;             // hidden features
    float* colsum = bufB + NF;             // [64]
    float* colsq  = colsum + 64;           // [64]
    float* scale  = colsq + 64;            // [64]
    float* shift  = scale + 64;            // [64]

    const int scatter_grid = (N_EDGES * 16) / 256;         // 62500
    const int gemm_grid_h  = ((N_NODES / 16) * 4 + 7) / 8; // 3125 (DOUT=64)
    const int gemm_grid_z  = ((N_NODES / 16) * 2 + 7) / 8; // 1563 (DOUT=32)
    const int act_grid     = (int)(NF / 4 / 256);          // 6250

    // ---- layer 1: propagate(x) @ W1 + b1 -> BN -> LeakyReLU ----
    hipMemsetAsync(bufA, 0, NF * sizeof(float), stream);
    hipMemsetAsync(colsum, 0, 128 * sizeof(float), stream);
    scatter_kernel<<<scatter_grid, 256, 0, stream>>>(x, ei, ew, bufA);
    gemm_wmma_kernel<64><<<gemm_grid_h, 256, 0, stream>>>(bufA, W1, b1, bufB, colsum, colsq);
    bn_finalize_kernel<<<1, 64, 0, stream>>>(colsum, colsq, g1, be1, scale, shift);
    bn_act_kernel<<<act_grid, 256, 0, stream>>>((float4*)bufB, (const float4*)scale, (const float4*)shift);

    // ---- layer 2 ----
    hipMemsetAsync(bufA, 0, NF * sizeof(float), stream);
    hipMemsetAsync(colsum, 0, 128 * sizeof(float), stream);
    scatter_kernel<<<scatter_grid, 256, 0, stream>>>(bufB, ei, ew, bufA);
    gemm_wmma_kernel<64><<<gemm_grid_h, 256, 0, stream>>>(bufA, W2, b2, bufB, colsum, colsq);
    bn_finalize_kernel<<<1, 64, 0, stream>>>(colsum, colsq, g2, be2, scale, shift);
    bn_act_kernel<<<act_grid, 256, 0, stream>>>((float4*)bufB, (const float4*)scale, (const float4*)shift);

    // ---- layer 3: propagate -> linear to D_Z=32, no BN / activation ----
    hipMemsetAsync(bufA, 0, NF * sizeof(float), stream);
    scatter_kernel<<<scatter_grid, 256, 0, stream>>>(bufB, ei, ew, bufA);
    gemm_wmma_kernel<32><<<gemm_grid_z, 256, 0, stream>>>(bufA, W3, b3, out, nullptr, nullptr);
}